// SimGNN_21680994910940
// MI455X (gfx1250) — compile-verified
//
#include <hip/hip_runtime.h>
#include <hip/hip_bf16.h>

typedef float v2f __attribute__((ext_vector_type(2)));
typedef float v8f __attribute__((ext_vector_type(8)));
typedef unsigned int u32x4 __attribute__((ext_vector_type(4)));
typedef int i32x4 __attribute__((ext_vector_type(4)));
typedef int i32x8 __attribute__((ext_vector_type(8)));

#define C_IN 32
#define C1   128
#define C2   64
#define C3   32
#define KDIM 16

// ---------------------------------------------------------------------------
// Tensor Data Mover: stage a contiguous array of f32 (nelem <= 65535) from
// global memory into LDS. D# per CDNA5 ISA §8.3/8.4:
//   group0: [1:0]=count=1 | lds_addr | global_addr[31:0] |
//           global_addr[56:32] + type=2 in [127:126]
//   group1: data_size=2 (4B); 1-D tile: tensor_dim0 = tile_dim0 = nelem,
//           tensor_dim1 = tile_dim1 = 1, stride0 = nelem. groups 2/3 zero.
// Tracked with TENSORcnt; issuing wave waits, then the workgroup barriers.
// This toolchain exposes the 6-arg builtin (g0, g1, g2, g3, g4, cpol).
// ---------------------------------------------------------------------------
__device__ __forceinline__ void tdm_load_f32(const void* gsrc, unsigned lds_off,
                                             unsigned nelem) {
  unsigned long long ga = (unsigned long long)gsrc;
  u32x4 g0;
  g0[0] = 1u;                                              // count=1
  g0[1] = lds_off;                                         // lds_addr (bytes)
  g0[2] = (unsigned)(ga & 0xFFFFFFFFu);                    // global_addr[31:0]
  g0[3] = (unsigned)((ga >> 32) & 0x01FFFFFFu) | 0x80000000u;  // [56:32] | type=2
  i32x8 g1;
  g1[0] = 0x00020000;                                      // data_size = 4 bytes
  g1[1] = (int)((nelem & 0xFFFFu) << 16);                  // tensor_dim0[15:0]
  g1[2] = (int)(((nelem >> 16) & 0xFFFFu) | (1u << 16));   // dim0[31:16] | dim1=1
  g1[3] = (int)((nelem & 0xFFFFu) << 16);                  // dim1 hi=0 | tile_dim0
  g1[4] = 1;                                               // tile_dim1=1, tile_dim2=0
  g1[5] = (int)nelem;                                      // tensor_dim0_stride lo32
  g1[6] = 0;
  g1[7] = 0;
  i32x4 z4 = {0, 0, 0, 0};
  i32x8 z8 = {0, 0, 0, 0, 0, 0, 0, 0};
  __builtin_amdgcn_tensor_load_to_lds(g0, g1, z4, z4, z8, 0);
}

// ---------------------------------------------------------------------------
// WMMA GEMM: H[nrows x COUT] = X[nrows x CIN] @ W[CIN x COUT], fp32,
// V_WMMA_F32_16X16X4_F32, one wave per 16x16 output tile. Whole W is staged
// into LDS once per workgroup via the TDM; B fragments come from ds_load.
// nrows must be a multiple of 16 (N = 100000 = 6250*16).
// ---------------------------------------------------------------------------
template <int CIN, int COUT>
__global__ void gemm_wmma_f32(const float* __restrict__ X,
                              const float* __restrict__ W,
                              float* __restrict__ H, int nrows) {
  static_assert(CIN * COUT <= 65535, "tile_dim0 is 16-bit");
  __shared__ float sW[CIN * COUT];

  if (threadIdx.x == 0) {
    tdm_load_f32(W, (unsigned)(size_t)sW, CIN * COUT);
    __builtin_amdgcn_s_wait_tensorcnt(0);
  }
  __syncthreads();

  const int wave = (int)((blockIdx.x * (unsigned)blockDim.x + threadIdx.x) >> 5);
  const int lane = threadIdx.x & 31;
  const int tilesN = COUT / 16;
  const int mt = wave / tilesN;           // wave-uniform
  const int nt = wave - mt * tilesN;
  if (mt * 16 >= nrows) return;           // wave-uniform -> EXEC all 1s inside
  const int m0 = mt * 16;
  const int n0 = nt * 16;
  const int half = lane >> 4;
  const int l = lane & 15;
  const float* __restrict__ xrow = X + (size_t)(m0 + l) * CIN;

  v8f acc = {};
#pragma unroll
  for (int kk = 0; kk < CIN; kk += 4) {
    v2f a, b;
    a.x = xrow[kk + 2 * half + 0];
    a.y = xrow[kk + 2 * half + 1];
    b.x = sW[(kk + 2 * half + 0) * COUT + n0 + l];
    b.y = sW[(kk + 2 * half + 1) * COUT + n0 + l];
    acc = __builtin_amdgcn_wmma_f32_16x16x4_f32(false, a, false, b,
                                                (short)0, acc, false, false);
  }
#pragma unroll
  for (int r = 0; r < 8; ++r)
    H[(size_t)(m0 + r + 8 * half) * COUT + n0 + l] = acc[r];
}

// ---------------------------------------------------------------------------
// Degree / normalization
// ---------------------------------------------------------------------------
__global__ void deg_init_kernel(float* __restrict__ deg, int n) {
  int i = blockIdx.x * blockDim.x + threadIdx.x;
  if (i < n) deg[i] = 1.0f;  // self-loop
}

__global__ void deg_edges_kernel(const int* __restrict__ dst,
                                 float* __restrict__ deg, int E) {
  int e = blockIdx.x * blockDim.x + threadIdx.x;
  if (e < E) atomicAdd(&deg[dst[e]], 1.0f);
}

__global__ void deg_rsqrt_kernel(float* __restrict__ deg, int n) {
  int i = blockIdx.x * blockDim.x + threadIdx.x;
  if (i < n) deg[i] = rsqrtf(deg[i]);  // deg >= 1 always
}

// ---------------------------------------------------------------------------
// GCN aggregation, factorized so the atomic loop is pure gather+add:
//   h'[n,c] = h[n,c]*dinv[n]           (scale pass, also zeroes agg)
//   agg[d,c] += h'[s,c]                (edge scatter, NO per-edge dinv work)
//   out[n,c] = (agg[n,c] + h'[n,c])*dinv[n] + b[c]   (finish; h' self-loop)
// ---------------------------------------------------------------------------
template <int C>
__global__ void scale_init_kernel(float* __restrict__ h,
                                  const float* __restrict__ dinv,
                                  float* __restrict__ agg, int n) {
  long long i = blockIdx.x * (long long)blockDim.x + threadIdx.x;
  long long total = (long long)n * C;
  if (i < total) {
    int node = (int)(i / C);
    h[i] *= dinv[node];
    agg[i] = 0.0f;
  }
}

template <int C>
__global__ void edge_scatter_kernel(const float* __restrict__ hs,
                                    const int* __restrict__ src,
                                    const int* __restrict__ dst,
                                    float* __restrict__ agg, int E) {
  const int G = C / 4;  // float4 groups per row (power of two)
  long long idx = blockIdx.x * (long long)blockDim.x + threadIdx.x;
  long long e = idx / G;
  int c = (int)(idx - e * G) * 4;
  if (e >= E) return;
  int s = src[e];
  int d = dst[e];
  const float4 v = *(const float4*)(hs + (size_t)s * C + c);
  float* ap = agg + (size_t)d * C + c;
  atomicAdd(ap + 0, v.x);
  atomicAdd(ap + 1, v.y);
  atomicAdd(ap + 2, v.z);
  atomicAdd(ap + 3, v.w);
}

template <int C, bool RELU>
__global__ void finish_kernel(const float* __restrict__ agg,
                              const float* __restrict__ hs,
                              const float* __restrict__ dinv,
                              const float* __restrict__ b,
                              float* __restrict__ x, int n) {
  long long i = blockIdx.x * (long long)blockDim.x + threadIdx.x;
  long long total = (long long)n * C;
  if (i < total) {
    int node = (int)(i / C);
    int c = (int)(i - (long long)node * C);
    float v = (agg[i] + hs[i]) * dinv[node] + b[c];
    x[i] = RELU ? fmaxf(v, 0.0f) : v;
  }
}

__global__ void zero32_kernel(float* __restrict__ p) {
  if (threadIdx.x < 32) p[threadIdx.x] = 0.0f;
}

// ---------------------------------------------------------------------------
// Attention pooling. Uses linearity: mean(x@attW) = mean(x)@attW.
// ---------------------------------------------------------------------------
__global__ void colsum_kernel(const float* __restrict__ x,
                              float* __restrict__ colsum, int nrows) {
  int lane = threadIdx.x & 31;
  int warp = (int)((blockIdx.x * (unsigned)blockDim.x + threadIdx.x) >> 5);
  int nwarps = (int)((gridDim.x * (unsigned)blockDim.x) >> 5);
  float acc = 0.0f;
  for (int n = warp; n < nrows; n += nwarps) acc += x[(size_t)n * C3 + lane];
  atomicAdd(&colsum[lane], acc);
}

__global__ void ctx_kernel(const float* __restrict__ colsum,
                           const float* __restrict__ attW,
                           float* __restrict__ ctx, int nrows) {
  int c = threadIdx.x;
  if (c < C3) {
    float inv = 1.0f / (float)nrows;
    float acc = 0.0f;
#pragma unroll
    for (int i = 0; i < C3; ++i) acc += (colsum[i] * inv) * attW[i * C3 + c];
    ctx[c] = tanhf(acc);
  }
}

// e[c] += sum_n x[n,c] * sigmoid(dot(x[n,:], ctx)); wave32 shuffle reduction.
__global__ void attpool_kernel(const float* __restrict__ x,
                               const float* __restrict__ ctx,
                               float* __restrict__ evec, int nrows) {
  int lane = threadIdx.x & 31;
  int warp = (int)((blockIdx.x * (unsigned)blockDim.x + threadIdx.x) >> 5);
  int nwarps = (int)((gridDim.x * (unsigned)blockDim.x) >> 5);
  float cw = ctx[lane];
  float acc = 0.0f;
  for (int n = warp; n < nrows; n += nwarps) {
    float v = x[(size_t)n * C3 + lane];
    float d = v * cw;
#pragma unroll
    for (int off = 16; off > 0; off >>= 1) d += __shfl_xor(d, off, 32);
    float sig = 1.0f / (1.0f + __expf(-d));
    acc += v * sig;
  }
  atomicAdd(&evec[lane], acc);
}

// ---------------------------------------------------------------------------
// NTN + MLP head (tiny, one block).
// ---------------------------------------------------------------------------
__global__ void ntn_head_kernel(const float* __restrict__ e1,
                                const float* __restrict__ e2,
                                const float* __restrict__ ntnW,
                                const float* __restrict__ ntnV,
                                const float* __restrict__ ntnb,
                                const float* __restrict__ fc1W,
                                const float* __restrict__ fc1b,
                                const float* __restrict__ outW,
                                const float* __restrict__ outb,
                                float* __restrict__ out) {
  __shared__ float s[KDIM];
  __shared__ float h[16];
  int t = threadIdx.x;
  if (t < KDIM) {
    float bil = 0.0f;
    for (int i = 0; i < C3; ++i) {
      float ei = e1[i];
      for (int j = 0; j < C3; ++j)
        bil += ei * ntnW[(size_t)(i * C3 + j) * KDIM + t] * e2[j];
    }
    float blk = 0.0f;
    for (int m = 0; m < C3; ++m) blk += ntnV[(size_t)t * (2 * C3) + m] * e1[m];
    for (int m = 0; m < C3; ++m) blk += ntnV[(size_t)t * (2 * C3) + C3 + m] * e2[m];
    float v = bil + blk + ntnb[t];
    s[t] = v > 0.0f ? v : 0.0f;
  }
  __syncthreads();
  if (t < 16) {
    float acc = fc1b[t];
#pragma unroll
    for (int k = 0; k < KDIM; ++k) acc += s[k] * fc1W[k * 16 + t];
    h[t] = acc > 0.0f ? acc : 0.0f;
  }
  __syncthreads();
  if (t == 0) {
    float acc = outb[0];
#pragma unroll
    for (int j = 0; j < 16; ++j) acc += h[j] * outW[j];
    out[0] = 1.0f / (1.0f + __expf(-acc));
  }
}

// ---------------------------------------------------------------------------
// Host side
// ---------------------------------------------------------------------------
static inline int nblk(long long n, int t) { return (int)((n + t - 1) / t); }

static void embed_graph(const float* feats, const int* src, const int* dst,
                        const float* W1, const float* b1,
                        const float* W2, const float* b2,
                        const float* W3, const float* b3,
                        const float* attW,
                        float* xbuf, float* hbuf, float* aggbuf, float* dinv,
                        float* colsum, float* ctx, float* evec,
                        int N, int E, hipStream_t stream) {
  const int T = 256;
  // degree -> dinv
  deg_init_kernel<<<nblk(N, T), T, 0, stream>>>(dinv, N);
  deg_edges_kernel<<<nblk(E, T), T, 0, stream>>>(dst, dinv, E);
  deg_rsqrt_kernel<<<nblk(N, T), T, 0, stream>>>(dinv, N);

  // layer 1: 32 -> 128, relu
  {
    long long waves = (long long)(N / 16) * (C1 / 16);
    gemm_wmma_f32<C_IN, C1><<<nblk(waves * 32, T), T, 0, stream>>>(feats, W1, hbuf, N);
    scale_init_kernel<C1><<<nblk((long long)N * C1, T), T, 0, stream>>>(hbuf, dinv, aggbuf, N);
    edge_scatter_kernel<C1><<<nblk((long long)E * (C1 / 4), T), T, 0, stream>>>(hbuf, src, dst, aggbuf, E);
    finish_kernel<C1, true><<<nblk((long long)N * C1, T), T, 0, stream>>>(aggbuf, hbuf, dinv, b1, xbuf, N);
  }
  // layer 2: 128 -> 64, relu
  {
    long long waves = (long long)(N / 16) * (C2 / 16);
    gemm_wmma_f32<C1, C2><<<nblk(waves * 32, T), T, 0, stream>>>(xbuf, W2, hbuf, N);
    scale_init_kernel<C2><<<nblk((long long)N * C2, T), T, 0, stream>>>(hbuf, dinv, aggbuf, N);
    edge_scatter_kernel<C2><<<nblk((long long)E * (C2 / 4), T), T, 0, stream>>>(hbuf, src, dst, aggbuf, E);
    finish_kernel<C2, true><<<nblk((long long)N * C2, T), T, 0, stream>>>(aggbuf, hbuf, dinv, b2, xbuf, N);
  }
  // layer 3: 64 -> 32, no relu; result into xbuf
  {
    long long waves = (long long)(N / 16) * (C3 / 16);
    gemm_wmma_f32<C2, C3><<<nblk(waves * 32, T), T, 0, stream>>>(xbuf, W3, hbuf, N);
    scale_init_kernel<C3><<<nblk((long long)N * C3, T), T, 0, stream>>>(hbuf, dinv, aggbuf, N);
    edge_scatter_kernel<C3><<<nblk((long long)E * (C3 / 4), T), T, 0, stream>>>(hbuf, src, dst, aggbuf, E);
    finish_kernel<C3, false><<<nblk((long long)N * C3, T), T, 0, stream>>>(aggbuf, hbuf, dinv, b3, xbuf, N);
  }
  // attention pooling on xbuf (N x 32)
  zero32_kernel<<<1, 32, 0, stream>>>(colsum);
  zero32_kernel<<<1, 32, 0, stream>>>(evec);
  colsum_kernel<<<1024, T, 0, stream>>>(xbuf, colsum, N);
  ctx_kernel<<<1, 32, 0, stream>>>(colsum, attW, ctx, N);
  attpool_kernel<<<1024, T, 0, stream>>>(xbuf, ctx, evec, N);
}

extern "C" void kernel_launch(void* const* d_in, const int* in_sizes, int n_in,
                              void* d_out, int out_size, void* d_ws, size_t ws_size,
                              hipStream_t stream) {
  const float* f1   = (const float*)d_in[0];
  const float* f2   = (const float*)d_in[1];
  const int*   eg1  = (const int*)d_in[2];
  const int*   eg2  = (const int*)d_in[3];
  const float* W1   = (const float*)d_in[4];
  const float* b1   = (const float*)d_in[5];
  const float* W2   = (const float*)d_in[6];
  const float* b2   = (const float*)d_in[7];
  const float* W3   = (const float*)d_in[8];
  const float* b3   = (const float*)d_in[9];
  const float* attW = (const float*)d_in[10];
  const float* ntnW = (const float*)d_in[11];
  const float* ntnV = (const float*)d_in[12];
  const float* ntnb = (const float*)d_in[13];
  const float* fc1W = (const float*)d_in[14];
  const float* fc1b = (const float*)d_in[15];
  const float* outW = (const float*)d_in[16];
  const float* outb = (const float*)d_in[17];

  const int N = in_sizes[0] / C_IN;
  const int E = in_sizes[2] / 2;

  float* ws = (float*)d_ws;
  const size_t nc = (size_t)N * C1;        // largest per-layer activation
  float* xbuf   = ws;
  float* hbuf   = xbuf + nc;
  float* aggbuf = hbuf + nc;
  float* dinv   = aggbuf + nc;
  float* colsum = dinv + N;
  float* ctx    = colsum + 32;
  float* ev1    = ctx + 32;
  float* ev2    = ev1 + 32;
  (void)ws_size; (void)n_in; (void)out_size;

  embed_graph(f1, eg1, eg1 + E, W1, b1, W2, b2, W3, b3, attW,
              xbuf, hbuf, aggbuf, dinv, colsum, ctx, ev1, N, E, stream);
  embed_graph(f2, eg2, eg2 + E, W1, b1, W2, b2, W3, b3, attW,
              xbuf, hbuf, aggbuf, dinv, colsum, ctx, ev2, N, E, stream);

  ntn_head_kernel<<<1, 32, 0, stream>>>(ev1, ev2, ntnW, ntnV, ntnb,
                                        fc1W, fc1b, outW, outb, (float*)d_out);
}